// GraphUnpool_53309134078318
// MI455X (gfx1250) — compile-verified
//
#include <hip/hip_runtime.h>

// Fixed problem shape from the reference.
#define B      8
#define N_OLD  2048
#define N_NEW  4096
#define FDIM   256

typedef float v4f __attribute__((ext_vector_type(4)));

// ---------------------------------------------------------------------------
// Kernel 1: zero the new_X region (B * N_NEW * FDIM floats) with B128 stores.
// Exact launch: n4 threads total, no bounds tail. RT policy: this region is
// immediately re-read by L2 atomics, keep it resident (32 MB << 192 MB L2).
// ---------------------------------------------------------------------------
__global__ void zero_newx(v4f* __restrict__ out4) {
    unsigned i = blockIdx.x * blockDim.x + threadIdx.x;
    out4[i] = (v4f){0.f, 0.f, 0.f, 0.f};
}

// ---------------------------------------------------------------------------
// Kernel 2: scatter-add X into new_X.
// One thread per float4 of X (B*N_OLD*FDIM/4 = 1,048,576 threads, exact grid).
// Duplicate idx values must sum -> native non-returning global_atomic_add_f32
// via unsafeAtomicAdd (confirmed by low VALU count / no CAS loop last round).
// X is stream-once -> non-temporal load.
// ---------------------------------------------------------------------------
__global__ void scatter_add(const v4f* __restrict__ X4,
                            const int* __restrict__ idx,
                            float*     __restrict__ newX) {
    const unsigned F4  = FDIM / 4u;                       // 64
    unsigned gid = blockIdx.x * blockDim.x + threadIdx.x; // [0, B*N_OLD*F4)
    unsigned row = gid / F4;                              // [0, B*N_OLD)
    unsigned f4  = gid - row * F4;
    unsigned b   = row / N_OLD;

    int node = idx[row];                                  // hot in L2 (64 threads share it)
    v4f x = __builtin_nontemporal_load(X4 + gid);

    float* dst = newX + ((size_t)b * N_NEW + (unsigned)node) * FDIM + f4 * 4u;
    unsafeAtomicAdd(dst + 0, x.x);
    unsafeAtomicAdd(dst + 1, x.y);
    unsafeAtomicAdd(dst + 2, x.z);
    unsafeAtomicAdd(dst + 3, x.w);
}

// ---------------------------------------------------------------------------
// Kernel 3: stream-copy A (536,870,912 bytes) via the gfx1250 async LDS path.
// Per phase each lane issues UNROLL async B128 loads global->LDS (NT), one
// s_wait_asynccnt 0, then UNROLL async B128 stores LDS->global (NT).
// Double-buffered across phases: the wait in phase i+1 retires the stores
// issued from the buffer that phase i+2 reuses. Each lane owns disjoint 16B
// LDS slices, so no cross-wave hazards and no barriers are needed.
// NT on both sides keeps the 512MB/dir stream from rinsing L2 (192 MB).
// ---------------------------------------------------------------------------
#define TILE_BYTES  4096u                 // 256 lanes * 16 B
#define UNROLL      4
#define GROUP_BYTES (TILE_BYTES * UNROLL) // 16 KB per block per phase

__global__ void copy_A_async(const float* __restrict__ src,
                             float*       __restrict__ dst,
                             unsigned num_groups) {
    __shared__ __align__(16) unsigned char smem[2][UNROLL][TILE_BYTES];
    const unsigned tid = threadIdx.x;

    // Flat address of a __shared__ object is {LDS aperture | group offset}:
    // low 32 bits are exactly the LDS byte offset the async ops need.
    const unsigned lds_base =
        (unsigned)(unsigned long long)(const void*)&smem[0][0][0] + tid * 16u;

    unsigned p = 0;
    for (unsigned g = blockIdx.x; g < num_groups; g += gridDim.x) {
        const unsigned goff  = g * GROUP_BYTES + tid * 16u;   // < 2^30, fits u32
        const unsigned ldsp  = lds_base + p * (UNROLL * TILE_BYTES);

#pragma unroll
        for (unsigned u = 0; u < UNROLL; ++u) {
            asm volatile("global_load_async_to_lds_b128 %0, %1, %2 th:TH_LOAD_NT"
                         :
                         : "v"(ldsp + u * TILE_BYTES),
                           "v"(goff + u * TILE_BYTES),
                           "s"(src)
                         : "memory");
        }
        // Retires this phase's loads AND the stores previously issued from
        // buffer p^1 in the last phase, which the next phase loads into.
        asm volatile("s_wait_asynccnt 0" ::: "memory");
#pragma unroll
        for (unsigned u = 0; u < UNROLL; ++u) {
            asm volatile("global_store_async_from_lds_b128 %0, %1, %2 th:TH_STORE_NT"
                         :
                         : "v"(goff + u * TILE_BYTES),
                           "v"(ldsp + u * TILE_BYTES),
                           "s"(dst)
                         : "memory");
        }
        p ^= 1;
    }
    asm volatile("s_wait_asynccnt 0" ::: "memory");
}

// ---------------------------------------------------------------------------
// Launch: d_in = {X, A, idx}; d_out = new_X (8,388,608 f32) ++ A (134,217,728 f32)
// ---------------------------------------------------------------------------
extern "C" void kernel_launch(void* const* d_in, const int* in_sizes, int n_in,
                              void* d_out, int out_size, void* d_ws, size_t ws_size,
                              hipStream_t stream) {
    (void)in_sizes; (void)n_in; (void)out_size; (void)d_ws; (void)ws_size;

    const float* X   = (const float*)d_in[0];
    const float* A   = (const float*)d_in[1];
    const int*   idx = (const int*)d_in[2];

    float* newX = (float*)d_out;
    float* outA = newX + (size_t)B * N_NEW * FDIM;       // +8,388,608

    // 1) zero new_X: 2,097,152 float4 -> 8192 blocks x 256 threads (exact)
    const unsigned n4 = (B * N_NEW * FDIM) / 4u;
    zero_newx<<<n4 / 256u, 256, 0, stream>>>((v4f*)newX);

    // 2) scatter-add: 1,048,576 float4 units -> 4096 blocks x 256 (exact)
    const unsigned nscat = B * N_OLD * (FDIM / 4u);
    scatter_add<<<nscat / 256u, 256, 0, stream>>>((const v4f*)X, idx, newX);

    // 3) async stream-copy of A: 32,768 groups of 16KB, grid-stride
    const unsigned groups =
        (unsigned)(((size_t)B * N_NEW * N_NEW * sizeof(float)) / GROUP_BYTES);
    copy_A_async<<<8192, 256, 0, stream>>>(A, outA, groups);
}